// DQ_Add_LayerNorm_Q_32968168964178
// MI455X (gfx1250) — compile-verified
//
#include <hip/hip_runtime.h>
#include <hip/hip_bf16.h>

#define DIM 4096
#define TPB 256                      // 8 wave32 waves per block
#define EPT (DIM / TPB)              // 16 elements per thread
#define NCHUNK (EPT / 4)             // 4 x float4 chunks per thread
#define INPUT_SCALE 0.01f
#define LN_EPS 1e-5f

typedef float v4f __attribute__((ext_vector_type(4)));
typedef int   v4i __attribute__((ext_vector_type(4)));

// address-space-qualified pointer types for the gfx1250 async-LDS builtin
typedef __attribute__((address_space(1))) v4i gas_v4i;   // global (printed as __device__)
typedef __attribute__((address_space(3))) v4i las_v4i;   // LDS

#if __has_builtin(__builtin_amdgcn_global_load_async_to_lds_b128)
#define HAVE_ASYNC_LDS 1
#else
#define HAVE_ASYNC_LDS 0
#endif

__global__ __launch_bounds__(TPB) void dq_add_ln_q_kernel(
    const float* __restrict__ residual,
    const int*   __restrict__ qin,
    const float* __restrict__ weight,
    const float* __restrict__ bias,
    float* __restrict__ out_x,
    float* __restrict__ out_q)
{
    // [0..DIM) = weight, [DIM..2*DIM) = bias  (32 KB of the 320 KB WGP LDS)
    __shared__ __align__(16) float s_wb[2 * DIM];
    __shared__ float s_red[16];      // 8 wave sums + 8 wave sums-of-squares

    const int tid  = threadIdx.x;
    const int lane = tid & 31;
    const int wave = tid >> 5;
    const size_t base = (size_t)blockIdx.x * (size_t)DIM;

    // ---- Stage weight||bias into LDS (CDNA5 async global->LDS, ASYNCcnt) ----
#if HAVE_ASYNC_LDS
    #pragma unroll
    for (int i = 0; i < 2 * NCHUNK; ++i) {
        const int lidx = i * (TPB * 4) + tid * 4;            // 0..8191 floats
        const float* gsrc = (i < NCHUNK) ? (weight + lidx) : (bias + (lidx - DIM));
        __builtin_amdgcn_global_load_async_to_lds_b128(
            (gas_v4i*)gsrc,
            (las_v4i*)(&s_wb[lidx]),
            0, 0);
    }
#else
    #pragma unroll
    for (int i = 0; i < 2 * NCHUNK; ++i) {
        const int lidx = i * (TPB * 4) + tid * 4;
        const float* gsrc = (i < NCHUNK) ? (weight + lidx) : (bias + (lidx - DIM));
        *(v4f*)(&s_wb[lidx]) = *(const v4f*)gsrc;
    }
#endif

    // ---- Pass 1: stream row, dequant+add, write x, accumulate moments ----
    float xv[EPT];
    float sum = 0.0f, sumsq = 0.0f;
    #pragma unroll
    for (int c = 0; c < NCHUNK; ++c) {
        const int    lcol = c * (TPB * 4) + tid * 4;
        const size_t col  = base + (size_t)lcol;
        v4f r = __builtin_nontemporal_load((const v4f*)(residual + col));
        v4i q = __builtin_nontemporal_load((const v4i*)(qin + col));
        v4f x;
        #pragma unroll
        for (int j = 0; j < 4; ++j) {
            const float xf = __builtin_fmaf((float)q[j], INPUT_SCALE, r[j]);
            x[j] = xf;
            xv[c * 4 + j] = xf;
            sum += xf;
            sumsq = __builtin_fmaf(xf, xf, sumsq);
        }
        __builtin_nontemporal_store(x, (v4f*)(out_x + col));
    }

    // ---- wave32 butterfly reduction, then cross-wave combine via LDS ----
    #pragma unroll
    for (int off = 16; off >= 1; off >>= 1) {
        sum   += __shfl_xor(sum,   off, 32);
        sumsq += __shfl_xor(sumsq, off, 32);
    }
    if (lane == 0) { s_red[wave] = sum; s_red[8 + wave] = sumsq; }

#if HAVE_ASYNC_LDS
    // retire this wave's async LDS writes before the workgroup barrier
  #if __has_builtin(__builtin_amdgcn_s_wait_asynccnt)
    __builtin_amdgcn_s_wait_asynccnt(0);
  #else
    asm volatile("s_wait_asynccnt 0" ::: "memory");
  #endif
#endif
    __syncthreads();

    float tot = 0.0f, totsq = 0.0f;
    #pragma unroll
    for (int w = 0; w < TPB / 32; ++w) { tot += s_red[w]; totsq += s_red[8 + w]; }
    const float invD = 1.0f / (float)DIM;
    const float mean = tot * invD;
    const float var  = __builtin_fmaf(-mean, mean, totsq * invD); // E[x^2]-mean^2
    const float rstd = rsqrtf(var + LN_EPS);

    // ---- Pass 2: normalize, affine (w/b from LDS), round-to-even, clamp ----
    #pragma unroll
    for (int c = 0; c < NCHUNK; ++c) {
        const int    lcol = c * (TPB * 4) + tid * 4;
        const size_t col  = base + (size_t)lcol;
        v4f w = *(const v4f*)(&s_wb[lcol]);
        v4f b = *(const v4f*)(&s_wb[DIM + lcol]);
        v4f o;
        #pragma unroll
        for (int j = 0; j < 4; ++j) {
            const float ln = __builtin_fmaf((xv[c * 4 + j] - mean) * rstd, w[j], b[j]);
            float qf = __builtin_rintf(ln);            // v_rndne_f32 (half-to-even)
            qf = fminf(fmaxf(qf, -128.0f), 127.0f);
            o[j] = qf;
        }
        __builtin_nontemporal_store(o, (v4f*)(out_q + col));
    }
}

extern "C" void kernel_launch(void* const* d_in, const int* in_sizes, int n_in,
                              void* d_out, int out_size, void* d_ws, size_t ws_size,
                              hipStream_t stream) {
    const float* residual = (const float*)d_in[0];
    const int*   qin      = (const int*)d_in[1];
    const float* weight   = (const float*)d_in[2];
    const float* bias     = (const float*)d_in[3];

    const int tokens = in_sizes[0] / DIM;          // 16384
    float* out_x = (float*)d_out;                  // fp32 residual stream out
    float* out_q = out_x + (size_t)tokens * DIM;   // quantized LN out (as float)

    dq_add_ln_q_kernel<<<tokens, TPB, 0, stream>>>(residual, qin, weight, bias,
                                                   out_x, out_q);
}